// MakeCutouts_44066364457610
// MI455X (gfx1250) — compile-verified
//
#include <hip/hip_runtime.h>
#include <math.h>

#define CUT  224
#define CUTN 64
#define NB   4
#define NC   3
#define IMH  512
#define IMW  512

// Per-cutout parameter block, produced by prep_kernel into d_ws.
struct KP {
  float H[9];                               // effective homography (identity if !persp)
  float ca, sa, inv_scale, tx, ty;          // affine params
  float sfac, sfm1;                         // s/224, s-1 for crop-resize
  int   oy, ox, s, ident;                   // crop window + identity-H flag
};

typedef __attribute__((ext_vector_type(2))) float v2f;
typedef __attribute__((ext_vector_type(8))) float v8f;

// ---------------------------------------------------------------------------
// Kernel 1: per-cutout prep (64 threads total). 8x8 Gauss-Jordan with partial
// pivoting for the homography, matching jnp.linalg.solve in fp32.
// ---------------------------------------------------------------------------
__global__ __launch_bounds__(64) void prep_kernel(
    const int*   __restrict__ sizes,
    const int*   __restrict__ offy,
    const int*   __restrict__ offx,
    const float* __restrict__ angles,
    const float* __restrict__ translates,
    const float* __restrict__ scales,
    const float* __restrict__ endpoints,   // (CUTN,4,2)
    const float* __restrict__ persp,
    KP*          __restrict__ kp)
{
  int k = threadIdx.x;
  if (k >= CUTN) return;

  KP p;
  int s = sizes[k];
  p.oy = offy[k]; p.ox = offx[k]; p.s = s;
  float sf = (float)s;
  p.sfac = sf / (float)CUT;
  p.sfm1 = sf - 1.0f;

  float rad = angles[k] * 0.017453292519943295f;  // pi/180
  p.ca = cosf(rad);
  p.sa = sinf(rad);
  p.inv_scale = 1.0f / scales[k];
  p.tx = translates[2 * k + 0];
  p.ty = translates[2 * k + 1];

  bool use_p = persp[k] > 0.5f;
  if (!use_p) {
    p.H[0] = 1.f; p.H[1] = 0.f; p.H[2] = 0.f;
    p.H[3] = 0.f; p.H[4] = 1.f; p.H[5] = 0.f;
    p.H[6] = 0.f; p.H[7] = 0.f; p.H[8] = 1.f;
    p.ident = 1;
  } else {
    // Build the 8x9 augmented system:
    //   rows q   : [x y 1 0 0 0 -xX -yX | X]
    //   rows q+4 : [0 0 0 x y 1 -xY -yY | Y]
    float M[8][9];
    const float Xc[4] = {0.f, 223.f, 223.f, 0.f};
    const float Yc[4] = {0.f, 0.f, 223.f, 223.f};
    for (int q = 0; q < 4; ++q) {
      float x = endpoints[k * 8 + q * 2 + 0];
      float y = endpoints[k * 8 + q * 2 + 1];
      float X = Xc[q], Y = Yc[q];
      M[q][0] = x;   M[q][1] = y;   M[q][2] = 1.f;
      M[q][3] = 0.f; M[q][4] = 0.f; M[q][5] = 0.f;
      M[q][6] = -x * X; M[q][7] = -y * X; M[q][8] = X;
      M[q+4][0] = 0.f; M[q+4][1] = 0.f; M[q+4][2] = 0.f;
      M[q+4][3] = x;   M[q+4][4] = y;   M[q+4][5] = 1.f;
      M[q+4][6] = -x * Y; M[q+4][7] = -y * Y; M[q+4][8] = Y;
    }
    // Gauss-Jordan with partial pivoting.
    for (int col = 0; col < 8; ++col) {
      int piv = col; float best = fabsf(M[col][col]);
      for (int r = col + 1; r < 8; ++r) {
        float v = fabsf(M[r][col]);
        if (v > best) { best = v; piv = r; }
      }
      if (piv != col) {
        for (int c = 0; c < 9; ++c) {
          float t = M[col][c]; M[col][c] = M[piv][c]; M[piv][c] = t;
        }
      }
      float inv = 1.0f / M[col][col];
      for (int r = 0; r < 8; ++r) {
        if (r == col) continue;
        float f = M[r][col] * inv;
        for (int c = 0; c < 9; ++c) M[r][c] -= f * M[col][c];
      }
    }
    for (int r = 0; r < 8; ++r) p.H[r] = M[r][8] / M[r][r];
    p.H[8] = 1.0f;
    p.ident = 0;
  }
  kp[k] = p;
}

// ---------------------------------------------------------------------------
// Fused sampling chain: crop-resize bilinear <- affine nearest <- persp tap.
// ---------------------------------------------------------------------------
__device__ __forceinline__ float sample_cut(const float* __restrict__ img,
                                            const KP& p, int yi, int xi) {
  // bilinear crop-resize of the s x s window at (oy,ox), output coord (yi,xi)
  float srcy = fminf(fmaxf(((float)yi + 0.5f) * p.sfac - 0.5f, 0.0f), p.sfm1);
  float srcx = fminf(fmaxf(((float)xi + 0.5f) * p.sfac - 0.5f, 0.0f), p.sfm1);
  float y0f = floorf(srcy), x0f = floorf(srcx);
  float wy = srcy - y0f,    wx = srcx - x0f;
  int y0 = (int)y0f, x0 = (int)x0f;
  int y1 = min(y0 + 1, p.s - 1);
  int x1 = min(x0 + 1, p.s - 1);
  const float* r0 = img + (size_t)(p.oy + y0) * IMW + p.ox;
  const float* r1 = img + (size_t)(p.oy + y1) * IMW + p.ox;
  float a = r0[x0], b = r0[x1], c = r1[x0], d = r1[x1];
  float top = a + wx * (b - a);
  float bot = c + wx * (d - c);
  return top + wy * (bot - top);
}

__device__ __forceinline__ float affine_pix(const float* __restrict__ img,
                                            const KP& p, int ii, int jj) {
  const float cx = (CUT - 1) * 0.5f;  // 111.5
  float xc = (float)jj - cx - p.tx;
  float yc = (float)ii - cx - p.ty;
  float xa = (p.ca * xc + p.sa * yc) * p.inv_scale + cx;
  float ya = (p.ca * yc - p.sa * xc) * p.inv_scale + cx;
  float xr = rintf(xa);   // v_rndne_f32: round-half-even == jnp.round
  float yr = rintf(ya);
  if (xr < 0.0f || xr > 223.0f || yr < 0.0f || yr > 223.0f) return 0.0f;
  return sample_cut(img, p, (int)yr, (int)xr);
}

__device__ __forceinline__ float persp_tap(const float* __restrict__ img,
                                           const KP& p, int yy, int xx) {
  if (xx < 0 || xx >= CUT || yy < 0 || yy >= CUT) return 0.0f;
  return affine_pix(img, p, yy, xx);
}

// ---------------------------------------------------------------------------
// Kernel 2: one wave32 per 16x16 output tile. Homogeneous coordinates
// (xnum, ynum, den) for the whole tile are produced by three
// V_WMMA_F32_16X16X4_F32 ops:  D[M,N] = (i0+M)*c_i + (j0+N)*c_j + c_0
// with A[M,:] = [i0+M, 1, 0, 0] and B = [c_i ; c_j*j+c_0 ; 0 ; 0].
// D layout: lane = tile column, VGPR r = tile row r (lanes 0-15) / r+8 (16-31).
// ---------------------------------------------------------------------------
__global__ __launch_bounds__(128) void warp_kernel(
    const float* __restrict__ input,
    const KP*    __restrict__ kps,
    float*       __restrict__ out)
{
  int plane = blockIdx.y;              // (k*NB + b)*NC + c, 0..767
  int k  = plane / (NB * NC);
  int bc = plane % (NB * NC);          // b*NC + c -> input channel plane
  const float* img = input + (size_t)bc * (IMH * IMW);
  KP p = kps[k];                       // uniform per block -> scalar path

  int wid  = threadIdx.x >> 5;
  int lane = threadIdx.x & 31;
  int t  = blockIdx.x * 4 + wid;       // tile 0..195 (14x14 tiles, exact)
  int i0 = (t / 14) * 16;
  int j0 = (t % 14) * 16;
  int col = lane & 15;
  bool hi = lane >= 16;
  int jg  = j0 + col;                  // this lane's output column
  float fj = (float)jg;

  // A matrix (16x4 f32): lane L<16 holds A[M=L,K=0..1] = [i0+L, 1];
  // lanes 16-31 hold K=2..3 which we zero.
  v2f a;
  a.x = hi ? 0.0f : (float)(i0 + lane);
  a.y = hi ? 0.0f : 1.0f;
  v8f z = {};

  // B matrices (4x16): row0 = c_i, row1 = c_j*j + c_0, rows 2-3 unused (A=0),
  // values placed to be correct under either documented B lane layout.
  v2f bx, by, bd;
  float tx_ = p.H[0] * fj + p.H[2];
  bx.x = hi ? tx_ : p.H[1];  bx.y = hi ? 0.0f : tx_;
  float ty_ = p.H[3] * fj + p.H[5];
  by.x = hi ? ty_ : p.H[4];  by.y = hi ? 0.0f : ty_;
  float td_ = p.H[6] * fj + p.H[8];
  bd.x = hi ? td_ : p.H[7];  bd.y = hi ? 0.0f : td_;

  v8f dx = __builtin_amdgcn_wmma_f32_16x16x4_f32(false, a, false, bx, (short)0, z, false, false);
  v8f dy = __builtin_amdgcn_wmma_f32_16x16x4_f32(false, a, false, by, (short)0, z, false, false);
  v8f dd = __builtin_amdgcn_wmma_f32_16x16x4_f32(false, a, false, bd, (short)0, z, false, false);

  int ibase = i0 + (hi ? 8 : 0);
  size_t obase = (size_t)plane * (CUT * CUT);

  if (p.ident) {
    // Identity homography: x_in=j, y_in=i exactly; single unit-weight tap.
#pragma unroll
    for (int r = 0; r < 8; ++r) {
      int i = ibase + r;
      out[obase + (size_t)i * CUT + jg] = affine_pix(img, p, i, jg);
    }
  } else {
#pragma unroll
    for (int r = 0; r < 8; ++r) {
      int i = ibase + r;
      float den  = dd[r];
      float x_in = dx[r] / den;
      float y_in = dy[r] / den;
      float x0f = floorf(x_in), y0f = floorf(y_in);
      float wx = x_in - x0f,    wy = y_in - y0f;
      int x0 = (int)x0f, y0 = (int)y0f;  // v_cvt_i32_f32 clamps wild values
      float v00 = persp_tap(img, p, y0,     x0);
      float v01 = persp_tap(img, p, y0,     x0 + 1);
      float v10 = persp_tap(img, p, y0 + 1, x0);
      float v11 = persp_tap(img, p, y0 + 1, x0 + 1);
      float val = v00 * (1.0f - wy) * (1.0f - wx)
                + v01 * (1.0f - wy) * wx
                + v10 * wy * (1.0f - wx)
                + v11 * wy * wx;
      out[obase + (size_t)i * CUT + jg] = val;
    }
  }
}

// ---------------------------------------------------------------------------
extern "C" void kernel_launch(void* const* d_in, const int* in_sizes, int n_in,
                              void* d_out, int out_size, void* d_ws, size_t ws_size,
                              hipStream_t stream) {
  const float* input      = (const float*)d_in[0];
  const int*   sizes      = (const int*)  d_in[1];
  const int*   offy       = (const int*)  d_in[2];
  const int*   offx       = (const int*)  d_in[3];
  const float* angles     = (const float*)d_in[4];
  const float* translates = (const float*)d_in[5];
  const float* scales     = (const float*)d_in[6];
  const float* endpoints  = (const float*)d_in[7];
  const float* persp      = (const float*)d_in[8];
  KP* kp = (KP*)d_ws;                       // 64 * 80 B = 5120 B of scratch
  float* out = (float*)d_out;

  prep_kernel<<<1, 64, 0, stream>>>(sizes, offy, offx, angles, translates,
                                    scales, endpoints, persp, kp);

  dim3 grid(49, CUTN * NB * NC);            // 49 blocks x 4 waves = 196 tiles; 768 planes
  warp_kernel<<<grid, 128, 0, stream>>>(input, kp, out);
}